// risk_35631048688255
// MI455X (gfx1250) — compile-verified
//
#include <hip/hip_runtime.h>
#include <math.h>

typedef __attribute__((ext_vector_type(2))) float v2f;
typedef __attribute__((ext_vector_type(8))) float v8f;

#define NTOT 2048
#define SEQ 32
#define DIM 256
#define USTRIDE 2064
#define MSTRIDE 1048576ull   /* floats per timestep slot (>= max nt*nc = 1024^2) */
#define K_ONE (1.0f + 1e-6f)

__device__ __forceinline__ unsigned int fkey(float f) {
  unsigned int u = __float_as_uint(f);
  return (f >= 0.0f) ? (u | 0x80000000u) : ~u;
}
__device__ __forceinline__ float funkey(unsigned int k) {
  return (k & 0x80000000u) ? __uint_as_float(k & 0x7fffffffu) : __uint_as_float(~k);
}
__device__ __forceinline__ float waveRedSum(float v) {
  for (int m = 16; m >= 1; m >>= 1) v += __shfl_xor(v, m, 32);
  return v;
}
__device__ __forceinline__ float waveRedMax(float v) {
  for (int m = 16; m >= 1; m >>= 1) v = fmaxf(v, __shfl_xor(v, m, 32));
  return v;
}

// ---- 1. Partition t into treated/control index lists; zero accumulators ----
__global__ void partition_kernel(const int* t, int* hdr, int* it, int* ic,
                                 unsigned int* maxkey, float* sumM, float* out) {
  int tid = threadIdx.x;
  if (tid < SEQ) { maxkey[tid] = 0u; sumM[tid] = 0.0f; }
  if (tid == 0) {
    int nt = 0, nc = 0;
    for (int i = 0; i < NTOT; ++i) {
      if (t[i] > 0) it[nt++] = i; else ic[nc++] = i;
    }
    hdr[0] = nt; hdr[1] = nc;
    ((float*)hdr)[2] = (float)nt / (float)NTOT;   // p = mean(t)
    out[0] = 0.0f;
  }
}

// ---- 2. Per (n,s) squared row norms, wave per row ----
__global__ void normsq_kernel(const float* __restrict__ enc, float* __restrict__ normsq) {
  int wavesPerBlock = blockDim.x >> 5;
  int gw = blockIdx.x * wavesPerBlock + (threadIdx.x >> 5);
  int lane = threadIdx.x & 31;
  int total = NTOT * SEQ;
  int stride = gridDim.x * wavesPerBlock;
  for (int w = gw; w < total; w += stride) {
    int n = w >> 5;       // / SEQ (SEQ==32)
    int s = w & 31;
    const float* p = enc + ((size_t)n * SEQ + s) * DIM;
    float acc = 0.0f;
    for (int j = 0; j < DIM; j += 32) {
      float x = p[j + lane];
      acc = fmaf(x, x, acc);
    }
    acc = waveRedSum(acc);
    if (lane == 0) normsq[s * NTOT + n] = acc;
  }
}

// ---- 3. M = ||xt||^2 + ||xc||^2 - 2 Xt Xc^T via fp32 WMMA 16x16x4; fused max/sum ----
__global__ void gemm_wmma_kernel(const float* __restrict__ enc, const int* hdr,
                                 const int* __restrict__ it, const int* __restrict__ ic,
                                 const float* __restrict__ normsq, float* __restrict__ Mbuf,
                                 unsigned int* maxkey, float* sumM) {
  int nt = hdr[0], nc = hdr[1];
  if (nt <= 0 || nc <= 0) return;
  int tilesR = (nt + 15) >> 4;
  int tilesC = (nc + 15) >> 4;
  long long total = (long long)SEQ * tilesR * tilesC;
  int wavesPerBlock = blockDim.x >> 5;
  int gw = blockIdx.x * wavesPerBlock + (threadIdx.x >> 5);
  int stride = gridDim.x * wavesPerBlock;
  int lane = threadIdx.x & 31;
  int half = lane >> 4;     // 0 or 1
  int m = lane & 15;
  for (long long tile = gw; tile < total; tile += stride) {
    int per = tilesR * tilesC;
    int s = (int)(tile / per);
    int tmp = (int)(tile - (long long)s * per);
    int tr = tmp / tilesC;
    int tc = tmp - tr * tilesC;
    int r0 = tr << 4, c0 = tc << 4;
    int ra = r0 + m; if (ra > nt - 1) ra = nt - 1;
    int ca = c0 + m; if (ca > nc - 1) ca = nc - 1;
    int rIdx = it[ra];
    int cIdx = ic[ca];
    const float* at = enc + ((size_t)rIdx * SEQ + s) * DIM;
    const float* bt = enc + ((size_t)cIdx * SEQ + s) * DIM;
    v8f acc = {};
    for (int k0 = 0; k0 < DIM; k0 += 4) {
      int k = k0 + 2 * half;
      v2f a; a.x = at[k]; a.y = at[k + 1];   // A: row m, K = k..k+1
      v2f b; b.x = bt[k]; b.y = bt[k + 1];   // B: col m, K = k..k+1
      acc = __builtin_amdgcn_wmma_f32_16x16x4_f32(false, a, false, b,
                                                  (short)0, acc, false, false);
    }
    float nsc = normsq[s * NTOT + cIdx];
    float lmax = -3.4e38f, lsum = 0.0f;
    float* Ms = Mbuf + (size_t)s * MSTRIDE;
    #pragma unroll
    for (int i = 0; i < 8; ++i) {
      int row = r0 + i + 8 * half;           // C layout: vgpr i -> M = i + 8*half
      int col = c0 + m;                      // N = lane%16
      if (row < nt && col < nc) {
        float nsr = normsq[s * NTOT + it[row]];
        float mval = fmaf(-2.0f, acc[i], nsr + nsc);
        Ms[(size_t)row * nc + col] = mval;
        lmax = fmaxf(lmax, mval);
        lsum += mval;
      }
    }
    lmax = waveRedMax(lmax);
    lsum = waveRedSum(lsum);
    if (lane == 0) {
      atomicMax(&maxkey[s], fkey(lmax));
      atomicAdd(&sumM[s], lsum);
    }
  }
}

// ---- 4a. Per-timestep scalars: delta, eff_lam, edge K value ----
__global__ void prep_kernel(const int* hdr, const unsigned int* maxkey,
                            const float* sumM, float* delta, float* lam, float* kd) {
  int s = threadIdx.x;
  if (s >= SEQ) return;
  int nt = hdr[0], nc = hdr[1];
  float d = funkey(maxkey[s]);
  float l = ((float)nt * (float)nc) / sumM[s];   // LAM / mean(M)
  delta[s] = d;
  lam[s] = l;
  kd[s] = __expf(-l * d) + 1e-6f;
}

// ---- 4b. u = a ----
__global__ void init_u_kernel(const int* hdr, float* u) {
  int nt = hdr[0];
  float p = ((const float*)hdr)[2];
  int total = SEQ * (nt + 1);
  for (int idx = blockIdx.x * blockDim.x + threadIdx.x; idx < total;
       idx += gridDim.x * blockDim.x) {
    int s = idx / (nt + 1);
    int r = idx - s * (nt + 1);
    u[s * USTRIDE + r] = (r < nt) ? (p / (float)nt) : (1.0f - p);
  }
}

// ---- 5. In-place K = exp(-lam * M) + 1e-6 ----
__global__ void exp_kernel(const int* hdr, const float* lam, float* Kbuf) {
  int nt = hdr[0], nc = hdr[1];
  if (nt <= 0 || nc <= 0) return;
  size_t nent = (size_t)nt * nc;
  size_t gstride = (size_t)gridDim.x * blockDim.x;
  size_t g0 = (size_t)blockIdx.x * blockDim.x + threadIdx.x;
  for (int s = 0; s < SEQ; ++s) {
    float l = lam[s];
    float* Ks = Kbuf + (size_t)s * MSTRIDE;
    for (size_t i = g0; i < nent; i += gstride) {
      Ks[i] = __expf(-l * Ks[i]) + 1e-6f;
    }
  }
}

// ---- 6a. Column pass: w = b / (K^T u); thread-per-column (coalesced), u in LDS ----
__global__ void colpass_kernel(const int* hdr, const float* __restrict__ Kbuf,
                               const float* kd, const float* __restrict__ u,
                               float* __restrict__ w) {
  __shared__ float uS[2080];
  int nt = hdr[0], nc = hdr[1];
  if (nt <= 0 || nc <= 0) return;
  float p = ((const float*)hdr)[2];
  float bc = (1.0f - p) / (float)nc;
  int cbcount = (nc + 255) >> 8;
  int total = SEQ * cbcount;
  int tid = threadIdx.x;
  int lane = tid & 31;
  for (int blk = blockIdx.x; blk < total; blk += gridDim.x) {
    int s = blk / cbcount;
    int cb = blk - s * cbcount;
    __syncthreads();
    for (int i = tid; i <= nt; i += blockDim.x) uS[i] = u[s * USTRIDE + i];
    __syncthreads();
    float kds = kd[s];
    int c = (cb << 8) + tid;
    if (c < nc) {
      const float* Kp = Kbuf + (size_t)s * MSTRIDE + c;
      float acc = 0.0f;
      int r = 0;
      for (; r + 4 <= nt; r += 4) {
        acc = fmaf(Kp[(size_t)r * nc],       uS[r],     acc);
        acc = fmaf(Kp[(size_t)(r + 1) * nc], uS[r + 1], acc);
        acc = fmaf(Kp[(size_t)(r + 2) * nc], uS[r + 2], acc);
        acc = fmaf(Kp[(size_t)(r + 3) * nc], uS[r + 3], acc);
      }
      for (; r < nt; ++r) acc = fmaf(Kp[(size_t)r * nc], uS[r], acc);
      float ktu = fmaf(kds, uS[nt], acc);          // padded row contributes kd*u[nt]
      w[s * USTRIDE + c] = bc / ktu;
    }
    if (cb == 0 && tid < 32) {                     // padded column element
      float su = 0.0f;
      for (int r = lane; r < nt; r += 32) su += uS[r];
      su = waveRedSum(su);
      if (lane == 0)
        w[s * USTRIDE + nc] = p / (kds * su + K_ONE * uS[nt]);
    }
  }
}

// ---- 6b. Row pass: u = a / (K w + kd*w[nc]); wave-per-row, w in LDS ----
__global__ void rowpass_kernel(const int* hdr, const float* __restrict__ Kbuf,
                               const float* kd, const float* __restrict__ w,
                               float* __restrict__ u) {
  __shared__ float wS[2080];
  int nt = hdr[0], nc = hdr[1];
  if (nt <= 0 || nc <= 0) return;
  float p = ((const float*)hdr)[2];
  float ar = p / (float)nt;
  int rowsPer = blockDim.x >> 5;
  int rbcount = (nt + 1 + rowsPer - 1) / rowsPer;
  int total = SEQ * rbcount;
  int tid = threadIdx.x;
  int lane = tid & 31;
  int wid = tid >> 5;
  for (int blk = blockIdx.x; blk < total; blk += gridDim.x) {
    int s = blk / rbcount;
    int rb = blk - s * rbcount;
    __syncthreads();
    for (int i = tid; i <= nc; i += blockDim.x) wS[i] = w[s * USTRIDE + i];
    __syncthreads();
    float kds = kd[s];
    int r = rb * rowsPer + wid;
    if (r < nt) {
      const float* Kp = Kbuf + (size_t)s * MSTRIDE + (size_t)r * nc;
      float acc = 0.0f;
      for (int c = lane; c < nc; c += 32) acc = fmaf(Kp[c], wS[c], acc);
      acc = waveRedSum(acc);
      if (lane == 0) u[s * USTRIDE + r] = ar / fmaf(kds, wS[nc], acc);
    } else if (r == nt) {                          // padded row of u
      float sw = 0.0f;
      for (int c = lane; c < nc; c += 32) sw += wS[c];
      sw = waveRedSum(sw);
      if (lane == 0) u[s * USTRIDE + nt] = (1.0f - p) / (kds * sw + K_ONE * wS[nc]);
    }
  }
}

// ---- 7. total += 2 * sum(T * Mt), M recomputed from K; plus padded-edge terms ----
__global__ void final_kernel(const int* hdr, const float* __restrict__ Kbuf,
                             const float* kd, const float* delta, const float* lam,
                             const float* __restrict__ u, const float* __restrict__ v,
                             float* out) {
  __shared__ float vS[2080];
  int nt = hdr[0], nc = hdr[1];
  if (nt <= 0 || nc <= 0) return;
  int rowsPer = blockDim.x >> 5;
  int rbcount = (nt + 1 + rowsPer - 1) / rowsPer;
  int total = SEQ * rbcount;
  int tid = threadIdx.x;
  int lane = tid & 31;
  int wid = tid >> 5;
  for (int blk = blockIdx.x; blk < total; blk += gridDim.x) {
    int s = blk / rbcount;
    int rb = blk - s * rbcount;
    __syncthreads();
    for (int i = tid; i <= nc; i += blockDim.x) vS[i] = v[s * USTRIDE + i];
    __syncthreads();
    float linv = 1.0f / lam[s];
    int r = rb * rowsPer + wid;
    if (r < nt) {
      const float* Kp = Kbuf + (size_t)s * MSTRIDE + (size_t)r * nc;
      float acc = 0.0f;
      for (int c = lane; c < nc; c += 32) {
        float kv = Kp[c];
        float mval = -__logf(fmaxf(kv - 1e-6f, 1e-30f)) * linv;
        acc += vS[c] * kv * mval;
      }
      acc = waveRedSum(acc);
      if (lane == 0) atomicAdd(out, 2.0f * u[s * USTRIDE + r] * acc);
    } else if (r == nt) {
      float sv = 0.0f, su = 0.0f;
      for (int c = lane; c < nc; c += 32) sv += vS[c];
      for (int rr = lane; rr < nt; rr += 32) su += u[s * USTRIDE + rr];
      sv = waveRedSum(sv);
      su = waveRedSum(su);
      if (lane == 0) {
        float edge = delta[s] * kd[s] * (vS[nc] * su + u[s * USTRIDE + nt] * sv);
        atomicAdd(out, 2.0f * edge);
      }
    }
  }
}

extern "C" void kernel_launch(void* const* d_in, const int* in_sizes, int n_in,
                              void* d_out, int out_size, void* d_ws, size_t ws_size,
                              hipStream_t stream) {
  const float* enc = (const float*)d_in[0];
  // d_in[1] = seq_len (==32, fixed)
  const int* t = (const int*)d_in[2];
  float* out = (float*)d_out;

  char* ws = (char*)d_ws;
  int* hdr = (int*)ws;                       ws += 16;          // nt, nc, p
  int* it = (int*)ws;                        ws += NTOT * 4;
  int* ic = (int*)ws;                        ws += NTOT * 4;
  unsigned int* maxkey = (unsigned int*)ws;  ws += SEQ * 4;
  float* sumM = (float*)ws;                  ws += SEQ * 4;
  float* delta = (float*)ws;                 ws += SEQ * 4;
  float* lam = (float*)ws;                   ws += SEQ * 4;
  float* kd = (float*)ws;                    ws += SEQ * 4;
  float* normsq = (float*)ws;                ws += (size_t)SEQ * NTOT * 4;
  float* u = (float*)ws;                     ws += (size_t)SEQ * USTRIDE * 4;
  float* w = (float*)ws;                     ws += (size_t)SEQ * USTRIDE * 4;
  float* Kbuf = (float*)ws;                  // SEQ * MSTRIDE floats (~128 MiB, L2-resident)

  partition_kernel<<<1, 64, 0, stream>>>(t, hdr, it, ic, maxkey, sumM, out);
  normsq_kernel<<<1024, 256, 0, stream>>>(enc, normsq);
  gemm_wmma_kernel<<<2048, 256, 0, stream>>>(enc, hdr, it, ic, normsq, Kbuf, maxkey, sumM);
  prep_kernel<<<1, 32, 0, stream>>>(hdr, maxkey, sumM, delta, lam, kd);
  init_u_kernel<<<64, 256, 0, stream>>>(hdr, u);
  exp_kernel<<<2048, 256, 0, stream>>>(hdr, lam, Kbuf);
  for (int i = 0; i < 20; ++i) {
    colpass_kernel<<<256, 256, 0, stream>>>(hdr, Kbuf, kd, u, w);
    rowpass_kernel<<<512, 256, 0, stream>>>(hdr, Kbuf, kd, w, u);
  }
  colpass_kernel<<<256, 256, 0, stream>>>(hdr, Kbuf, kd, u, w);   // w becomes v
  final_kernel<<<512, 256, 0, stream>>>(hdr, Kbuf, kd, delta, lam, u, w, out);
}